// TCKTNet_62938450756005
// MI455X (gfx1250) — compile-verified
//
#include <hip/hip_runtime.h>
#include <hip/hip_bf16.h>
#include <math.h>

// ---------------- sizes ----------------
#define BB 16
#define TT 500
#define DK 128
#define DH 16
#define HH 8
#define NC 101           // N_C + 1
#define NGLOB 2000
#define NEGV (-1e9f)

typedef __attribute__((ext_vector_type(2))) float v2f;
typedef __attribute__((ext_vector_type(4))) float v4f;
typedef __attribute__((ext_vector_type(8))) float v8f;

// D = A(16x4) * B(4x16) + C, fp32 WMMA on gfx1250
__device__ inline v8f wmma4(v2f a, v2f b, v8f c) {
  return __builtin_amdgcn_wmma_f32_16x16x4_f32(false, a, false, b, (short)0, c,
                                               false, false);
}

// lane half 0 takes elements {0,1}, half 1 takes {2,3} of the 4-wide K chunk
__device__ inline v2f pick2(v4f q, int half) {
  v2f r;
  r[0] = half ? q[2] : q[0];
  r[1] = half ? q[3] : q[1];
  return r;
}

__device__ inline float sigm(float x) { return 1.0f / (1.0f + expf(-x)); }

// =====================================================================
// Generic GEMM: C[m][n] = sum_k A(m,k) * W[n][k] + bias[n]
// A is a 2-way concat: k < K1 -> A1, else A2.  One wave = one 16x16 tile.
// M, N multiples of 16; K1, K2 multiples of 4 (16B-aligned rows).
// =====================================================================
__global__ __launch_bounds__(256) void gemm_cat_wmma(
    const float* __restrict__ A1, int K1, const float* __restrict__ A2, int K2,
    const float* __restrict__ W, const float* __restrict__ bias,
    float* __restrict__ C, int M, int N) {
  const int lane = threadIdx.x & 31;
  const int wv = threadIdx.x >> 5;
  const int tilesN = N >> 4;
  const int gid = blockIdx.x * 8 + wv;
  const int totalTiles = (M >> 4) * tilesN;
  if (gid >= totalTiles) return;
  const int tm = gid / tilesN, tn = gid % tilesN;
  const int half = lane >> 4, l15 = lane & 15;
  const int mrow = tm * 16 + l15;
  const int ncol = tn * 16 + l15;
  const float* __restrict__ a1row = A1 + (size_t)mrow * K1;
  const float* __restrict__ a2row = A2 + (size_t)mrow * K2;
  const float* __restrict__ wrow = W + (size_t)ncol * (K1 + K2);
  v8f c = (v8f)0.0f;
#pragma unroll 4
  for (int k = 0; k < K1; k += 4) {
    const v4f a4 = *(const v4f*)(a1row + k);
    const v4f w4 = *(const v4f*)(wrow + k);
    c = wmma4(pick2(a4, half), pick2(w4, half), c);
  }
#pragma unroll 4
  for (int k = 0; k < K2; k += 4) {
    const v4f a4 = *(const v4f*)(a2row + k);
    const v4f w4 = *(const v4f*)(wrow + K1 + k);
    c = wmma4(pick2(a4, half), pick2(w4, half), c);
  }
  const float bv = bias[ncol];
  for (int v = 0; v < 8; ++v)
    C[(size_t)(tm * 16 + v + 8 * half) * N + ncol] = c[v] + bv;
}

// =====================================================================
// Fused gather-GEMM for all0 = W_all @ [e_emb | c_emb | a_rep | d_rep] + b
// M = 8000 (rows b*T+t), K = 384, N = 128.  Output -> save_all region.
// =====================================================================
__global__ __launch_bounds__(256) void gemm_all0_wmma(
    const float* __restrict__ E_e, const float* __restrict__ E_c,
    const int* __restrict__ e_idx, const int* __restrict__ c_idx,
    const int* __restrict__ a_dat, const float* __restrict__ ediff,
    const float* __restrict__ W, const float* __restrict__ bias,
    float* __restrict__ C) {
  const int lane = threadIdx.x & 31;
  const int wv = threadIdx.x >> 5;
  const int gid = blockIdx.x * 8 + wv;
  const int totalTiles = (BB * TT / 16) * (DK / 16);  // 500*8
  if (gid >= totalTiles) return;
  const int tilesN = DK / 16;
  const int tm = gid / tilesN, tn = gid % tilesN;
  const int half = lane >> 4, l15 = lane & 15;
  const int mrow = tm * 16 + l15;
  const int ncol = tn * 16 + l15;
  const float* __restrict__ erow = E_e + (size_t)e_idx[mrow] * DK;
  const float* __restrict__ crow = E_c + (size_t)c_idx[mrow] * DK;
  const float av = (float)a_dat[mrow];
  const float dv = ediff[mrow];
  const float* __restrict__ wrow = W + (size_t)ncol * 384;
  v8f c = (v8f)0.0f;
#pragma unroll 4
  for (int k = 0; k < 128; k += 4) {  // e_emb region
    const v4f a4 = *(const v4f*)(erow + k);
    const v4f w4 = *(const v4f*)(wrow + k);
    c = wmma4(pick2(a4, half), pick2(w4, half), c);
  }
#pragma unroll 4
  for (int k = 0; k < 128; k += 4) {  // c_emb region
    const v4f a4 = *(const v4f*)(crow + k);
    const v4f w4 = *(const v4f*)(wrow + 128 + k);
    c = wmma4(pick2(a4, half), pick2(w4, half), c);
  }
  v2f aa;
  aa[0] = av;
  aa[1] = av;
#pragma unroll 4
  for (int k = 0; k < 64; k += 4) {  // a_rep region (splat)
    const v4f w4 = *(const v4f*)(wrow + 256 + k);
    c = wmma4(aa, pick2(w4, half), c);
  }
  v2f dd;
  dd[0] = dv;
  dd[1] = dv;
#pragma unroll 4
  for (int k = 0; k < 64; k += 4) {  // d_rep region (splat)
    const v4f w4 = *(const v4f*)(wrow + 320 + k);
    c = wmma4(dd, pick2(w4, half), c);
  }
  const float bv = bias[ncol];
  for (int v = 0; v < 8; ++v)
    C[(size_t)(tm * 16 + v + 8 * half) * DK + ncol] = c[v] + bv;
}

// =====================================================================
// X[(t*B+b)*128+d] = save_all[(b*T+t)*128+d] + pos_is[t*128+d]
// =====================================================================
__global__ void build_x_kernel(const float* __restrict__ save,
                               const float* __restrict__ pos,
                               float* __restrict__ X) {
  const int i = blockIdx.x * blockDim.x + threadIdx.x;
  if (i >= BB * TT * DK) return;
  const int d = i & (DK - 1);
  const int r = i >> 7;  // t*B + b
  const int t = r >> 4, b = r & 15;
  X[i] = save[((size_t)b * TT + t) * DK + d] + pos[(size_t)t * DK + d];
}

// =====================================================================
// out = LayerNorm(a + res) * g + be ; rows of 128, one wave per row
// =====================================================================
__global__ __launch_bounds__(256) void add_ln_kernel(
    const float* __restrict__ a, const float* __restrict__ res,
    const float* __restrict__ g, const float* __restrict__ be,
    float* __restrict__ out, int rows) {
  const int row = blockIdx.x * 8 + (threadIdx.x >> 5);
  if (row >= rows) return;
  const int lane = threadIdx.x & 31;
  float x[4];
  float s = 0.0f;
  for (int i = 0; i < 4; ++i) {
    const int d = lane + 32 * i;
    x[i] = a[(size_t)row * DK + d] + res[(size_t)row * DK + d];
    s += x[i];
  }
  for (int m = 16; m >= 1; m >>= 1) s += __shfl_xor(s, m);
  const float mean = s * (1.0f / DK);
  float vs = 0.0f;
  for (int i = 0; i < 4; ++i) {
    const float dd = x[i] - mean;
    vs += dd * dd;
  }
  for (int m = 16; m >= 1; m >>= 1) vs += __shfl_xor(vs, m);
  const float inv = rsqrtf(vs * (1.0f / DK) + 1e-5f);
  for (int i = 0; i < 4; ++i) {
    const int d = lane + 32 * i;
    out[(size_t)row * DK + d] = (x[i] - mean) * inv * g[d] + be[d];
  }
}

// =====================================================================
// Flash attention, wave per (b, h, q-tile).  Computes S^T = K x Q^T so
// the softmax state is per-lane (query = lane&15), then O^T = V^T x P^T.
// Q/O rows at (l*B + b)*128; K/V row k at K + k*kRow + b*kBatch.
// Causal: key k allowed iff k <= q (matches triu(...,1) masks).
// =====================================================================
__global__ __launch_bounds__(256) void attn_wmma_kernel(
    const float* __restrict__ Q, const float* __restrict__ K,
    const float* __restrict__ V, float* __restrict__ O, int Lq, int Lk,
    int kRow, int kBatch, int nTilesQ) {
  const int lane = threadIdx.x & 31;
  const int wid = blockIdx.x * 8 + (threadIdx.x >> 5);
  const int total = BB * HH * nTilesQ;
  if (wid >= total) return;
  const int qt = wid % nTilesQ;
  const int bh = wid / nTilesQ;
  const int h = bh % HH;
  const int b = bh / HH;
  const int half = lane >> 4, n = lane & 15;
  const int q0 = qt * 16;
  const int qrow = q0 + n;                        // this lane's query
  const int qgl = (qrow < Lq) ? qrow : (Lq - 1);  // clamped for loads
  const float* __restrict__ qptr = Q + ((size_t)qgl * BB + b) * DK + h * DH;
  const float* __restrict__ Kb = K + (size_t)b * kBatch + h * DH;
  const float* __restrict__ Vb = V + (size_t)b * kBatch + h * DH;
  v2f qf[4];
#pragma unroll
  for (int s = 0; s < 4; ++s) {
    const v4f q4 = *(const v4f*)(qptr + 4 * s);
    qf[s] = pick2(q4, half);
  }
  float m_i = -3.0e38f, l_i = 0.0f;
  v8f ot = (v8f)0.0f;
  int kmax = q0 + 15;
  if (kmax >= Lk) kmax = Lk - 1;
  const int nkt = (kmax >> 4) + 1;
  for (int kt = 0; kt < nkt; ++kt) {
    // ---- scores: St(16k x 16q) = K_tile x Q^T, dh = 16 -> 4 wmma steps
    const int kr = kt * 16 + n;
    const int krc = (kr < Lk) ? kr : 0;
    const float* __restrict__ kptr = Kb + (size_t)krc * kRow;
    v8f st = (v8f)0.0f;
#pragma unroll
    for (int s = 0; s < 4; ++s) {
      const v4f k4 = *(const v4f*)(kptr + 4 * s);
      st = wmma4(pick2(k4, half), qf[s], st);
    }
    // ---- mask + online softmax (query = lane&15, halves replicated)
    float p[8];
    float tmax = -3.0e38f;
    for (int v = 0; v < 8; ++v) {
      const int kk = kt * 16 + v + 8 * half;
      float sv = st[v] * 0.25f;  // 1/sqrt(16)
      if (kk > qrow || kk >= Lk) sv = NEGV;
      p[v] = sv;
      tmax = fmaxf(tmax, sv);
    }
    tmax = fmaxf(tmax, __shfl_xor(tmax, 16));
    const float mnew = fmaxf(m_i, tmax);
    const float scale = expf(m_i - mnew);
    float lsum = 0.0f;
    for (int v = 0; v < 8; ++v) {
      p[v] = expf(p[v] - mnew);
      lsum += p[v];
    }
    lsum += __shfl_xor(lsum, 16);
    l_i = l_i * scale + lsum;
    m_i = mnew;
    for (int v = 0; v < 8; ++v) ot[v] *= scale;
    float po[8];
    for (int v = 0; v < 8; ++v) po[v] = __shfl_xor(p[v], 16);
    // ---- Ot(16d x 16q) += V^T x Pt
#pragma unroll
    for (int s = 0; s < 4; ++s) {
      v2f a, pb;
      for (int j = 0; j < 2; ++j) {
        const int kk = kt * 16 + 4 * s + 2 * half + j;
        const int kc = (kk < Lk) ? kk : 0;  // p ~ 0 there anyway
        a[j] = Vb[(size_t)kc * kRow + n];
        const int kloc = 4 * s + 2 * half + j;
        pb[j] = ((kloc >> 3) == half) ? p[kloc & 7] : po[kloc & 7];
      }
      ot = wmma4(a, pb, ot);
    }
  }
  if (qrow < Lq) {
    const float inv = 1.0f / l_i;
    float* __restrict__ optr = O + ((size_t)qrow * BB + b) * DK + h * DH;
    for (int v = 0; v < 8; ++v) optr[v + 8 * half] = ot[v] * inv;
  }
}

// =====================================================================
// Persistent scan kernel: one block, 256 threads (8 waves).
// Gate GEMMs (16x128, K=256) done with WMMA: wave w computes cols 16w..
// =====================================================================
__device__ inline void wave_gemm16(const float* __restrict__ s_in,
                                   const float* __restrict__ W,
                                   const float* __restrict__ bias,
                                   float* __restrict__ s_out, int wv,
                                   int lane) {
  const int half = lane >> 4, l15 = lane & 15;
  const int n = wv * 16 + l15;
  const float* __restrict__ wrow = W + (size_t)n * 256;
  const float* __restrict__ arow = s_in + l15 * 256;
  v8f c = (v8f)0.0f;
#pragma unroll 4
  for (int k = 0; k < 256; k += 4) {
    const v4f a4 = *(const v4f*)(arow + k);   // ds_load_b128
    const v4f w4 = *(const v4f*)(wrow + k);   // global_load_b128
    c = wmma4(pick2(a4, half), pick2(w4, half), c);
  }
  const float bv = bias[n];
  for (int v = 0; v < 8; ++v) s_out[(v + 8 * half) * DK + n] = c[v] + bv;
}

__global__ __launch_bounds__(256) void scan_kernel(
    const float* __restrict__ ALL_L, const float* __restrict__ E_at,
    const float* __restrict__ E_it, const float* __restrict__ E_e,
    const int* __restrict__ at_data, const int* __restrict__ it_data,
    const int* __restrict__ e_data, const float* __restrict__ q_matrix,
    const float* __restrict__ W_lat, const float* __restrict__ b_lat,
    const float* __restrict__ W_lath, const float* __restrict__ b_lath,
    const float* __restrict__ W_lit, const float* __restrict__ b_lit,
    const float* __restrict__ W_lith, const float* __restrict__ b_lith,
    const float* __restrict__ W_y, const float* __restrict__ b_y,
    const float* __restrict__ h0, float* __restrict__ H,
    float* __restrict__ pred) {
  __shared__ float s_in[16 * 256];
  __shared__ float s_tmp[16 * DK];
  __shared__ float s_lt[16 * DK];
  __shared__ float s_gf[16 * DK];
  __shared__ float s_ht[16 * DK];
  __shared__ float s_qt[16 * NC];
  __shared__ float s_qn[16 * NC];
  const int tid = threadIdx.x;
  const int lane = tid & 31, wv = tid >> 5;

  // ---- init: H = broadcast h0, ht0 = qe[:,0] @ h0, pred[:,0] = 0
  for (int i = tid; i < 16 * NC * DK; i += 256) H[i] = h0[i % (NC * DK)];
  for (int i = tid; i < 16 * NC; i += 256) {
    const int b = i / NC, c = i % NC;
    s_qt[i] = q_matrix[(size_t)e_data[b * TT + 0] * NC + c];
  }
  __syncthreads();
  for (int i = tid; i < 16 * DK; i += 256) {
    const int b = i / DK, d = i % DK;
    float acc = 0.0f;
    for (int c = 0; c < NC; ++c) acc += s_qt[b * NC + c] * h0[c * DK + d];
    s_ht[i] = acc;
  }
  if (tid < 16) pred[tid * TT + 0] = 0.0f;
  __syncthreads();

  for (int t = 0; t < TT - 1; ++t) {
    // phase A: s_in = [lrn | at_emb]; gather q_t, q_n
    for (int i = tid; i < 16 * DK; i += 256) {
      const int b = i / DK, d = i % DK;
      s_in[b * 256 + d] = ALL_L[((size_t)t * 16 + b) * DK + d];
      s_in[b * 256 + 128 + d] = E_at[(size_t)at_data[b * TT + t] * DK + d];
    }
    for (int i = tid; i < 16 * NC; i += 256) {
      const int b = i / NC, c = i % NC;
      s_qt[i] = q_matrix[(size_t)e_data[b * TT + t] * NC + c];
      s_qn[i] = q_matrix[(size_t)e_data[b * TT + t + 1] * NC + c];
    }
    __syncthreads();
    wave_gemm16(s_in, W_lat, b_lat, s_tmp, wv, lane);  // la1
    __syncthreads();
    for (int i = tid; i < 16 * DK; i += 256) {
      const int b = i / DK, d = i % DK;
      s_in[b * 256 + d] = s_ht[i];
      s_in[b * 256 + 128 + d] = s_tmp[i];
    }
    __syncthreads();
    wave_gemm16(s_in, W_lath, b_lath, s_lt, wv, lane);  // la (raw)
    __syncthreads();
    for (int i = tid; i < 16 * DK; i += 256) {
      const float la = s_lt[i];
      s_lt[i] = sigm(la) * (tanhf(la) + 1.0f) * 0.5f;
    }
    __syncthreads();
    for (int i = tid; i < 16 * DK; i += 256) {
      const int b = i / DK, d = i % DK;
      s_in[b * 256 + d] = ALL_L[((size_t)t * 16 + b) * DK + d];
      s_in[b * 256 + 128 + d] = E_it[(size_t)it_data[b * TT + t] * DK + d];
    }
    __syncthreads();
    wave_gemm16(s_in, W_lit, b_lit, s_tmp, wv, lane);  // li1
    __syncthreads();
    for (int i = tid; i < 16 * DK; i += 256) {
      const int b = i / DK, d = i % DK;
      s_in[b * 256 + d] = s_ht[i];
      s_in[b * 256 + 128 + d] = s_tmp[i];
    }
    __syncthreads();
    wave_gemm16(s_in, W_lith, b_lith, s_gf, wv, lane);  // li (raw)
    __syncthreads();
    for (int i = tid; i < 16 * DK; i += 256) s_gf[i] = sigm(s_gf[i]);
    __syncthreads();
    // h update + ht_new = sum_c q_n * h_new
    for (int i = tid; i < 16 * DK; i += 256) {
      const int b = i / DK, d = i % DK;
      const float gf = s_gf[i], lt = s_lt[i];
      float acc = 0.0f;
      float* __restrict__ hp = H + (size_t)b * NC * DK + d;
      const float* __restrict__ qtp = s_qt + b * NC;
      const float* __restrict__ qnp = s_qn + b * NC;
      for (int c = 0; c < NC; ++c) {
        const float hv = hp[(size_t)c * DK] * gf + qtp[c] * lt;
        hp[(size_t)c * DK] = hv;
        acc += qnp[c] * hv;
      }
      s_tmp[i] = acc;
    }
    __syncthreads();
    for (int i = tid; i < 16 * DK; i += 256) {
      const int b = i / DK, d = i % DK;
      const float htn = s_tmp[i];
      s_ht[i] = htn;
      s_in[b * 256 + d] = E_e[(size_t)e_data[b * TT + t + 1] * DK + d];
      s_in[b * 256 + 128 + d] = htn;
    }
    __syncthreads();
    wave_gemm16(s_in, W_y, b_y, s_tmp, wv, lane);  // y logits
    __syncthreads();
    if (tid < 16) {
      float acc = 0.0f;
      for (int d = 0; d < DK; ++d) acc += sigm(s_tmp[tid * DK + d]);
      pred[tid * TT + (t + 1)] = acc * (1.0f / DK);
    }
    __syncthreads();
  }
}

// =====================================================================
// launcher
// =====================================================================
extern "C" void kernel_launch(void* const* d_in, const int* in_sizes, int n_in,
                              void* d_out, int out_size, void* d_ws,
                              size_t ws_size, hipStream_t stream) {
  // --- params (depth-first setup_inputs() dict order) ---
  const float* E_at = (const float*)d_in[0];
  const float* E_it = (const float*)d_in[1];
  const float* E_e = (const float*)d_in[2];
  const float* E_c = (const float*)d_in[3];
  const float* W_all = (const float*)d_in[4];
  const float* b_all = (const float*)d_in[5];
  const float* pos_is = (const float*)d_in[6];
  const float* Wv_is = (const float*)d_in[7];
  const float* bv_is = (const float*)d_in[8];
  const float* Wk_is = (const float*)d_in[9];
  const float* bk_is = (const float*)d_in[10];
  const float* Wq_is = (const float*)d_in[11];
  const float* bq_is = (const float*)d_in[12];
  const float* is_wq = (const float*)d_in[13];
  const float* is_bq = (const float*)d_in[14];
  const float* is_wk = (const float*)d_in[15];
  const float* is_bk = (const float*)d_in[16];
  const float* is_wv = (const float*)d_in[17];
  const float* is_bv = (const float*)d_in[18];
  const float* is_wo = (const float*)d_in[19];
  const float* is_bo = (const float*)d_in[20];
  const float* g_is = (const float*)d_in[21];
  const float* be_is = (const float*)d_in[22];
  const float* MC = (const float*)d_in[23];
  const float* Wv_cs = (const float*)d_in[24];
  const float* bv_cs = (const float*)d_in[25];
  const float* Wk_cs = (const float*)d_in[26];
  const float* bk_cs = (const float*)d_in[27];
  const float* Wq_cs = (const float*)d_in[28];
  const float* bq_cs = (const float*)d_in[29];
  const float* cs_wq = (const float*)d_in[30];
  const float* cs_bq = (const float*)d_in[31];
  const float* cs_wk = (const float*)d_in[32];
  const float* cs_bk = (const float*)d_in[33];
  const float* cs_wv = (const float*)d_in[34];
  const float* cs_bv = (const float*)d_in[35];
  const float* cs_wo = (const float*)d_in[36];
  const float* cs_bo = (const float*)d_in[37];
  const float* g_cs = (const float*)d_in[38];
  const float* be_cs = (const float*)d_in[39];
  const float* W_fc = (const float*)d_in[40];
  const float* b_fc = (const float*)d_in[41];
  const float* W_lat = (const float*)d_in[42];
  const float* b_lat = (const float*)d_in[43];
  const float* W_lath = (const float*)d_in[44];
  const float* b_lath = (const float*)d_in[45];
  const float* W_lit = (const float*)d_in[46];
  const float* b_lit = (const float*)d_in[47];
  const float* W_lith = (const float*)d_in[48];
  const float* b_lith = (const float*)d_in[49];
  const float* W_y = (const float*)d_in[50];
  const float* b_y = (const float*)d_in[51];
  const float* h0 = (const float*)d_in[52];
  // --- non-param inputs ---
  const float* q_matrix = (const float*)d_in[53];
  const int* e_data = (const int*)d_in[54];
  const int* at_data = (const int*)d_in[55];
  const int* a_data = (const int*)d_in[56];
  const int* it_data = (const int*)d_in[57];
  const int* c_data = (const int*)d_in[58];
  // d_in[59] ca_data, d_in[60] recent_c: unused (matches reference)
  const float* e_diff = (const float*)d_in[61];

  float* pred = (float*)d_out;   // (16,500)
  float* save = pred + BB * TT;  // (8000,128) = save_all

  float* ws = (float*)d_ws;
  const size_t SZ = (size_t)BB * TT * DK;  // 1,024,000 floats
  float* X = ws + 0 * SZ;
  float* QQ = ws + 1 * SZ;  // reused as QH in cs block
  float* KQ = ws + 2 * SZ;  // reused as KH
  float* VQ = ws + 3 * SZ;  // reused as VH
  float* Q2 = ws + 4 * SZ;
  float* K2 = ws + 5 * SZ;
  float* V2 = ws + 6 * SZ;
  float* CTX = ws + 7 * SZ;
  float* MHO = ws + 8 * SZ;
  float* AIS = ws + 9 * SZ;
  float* ACS = ws + 10 * SZ;
  float* ALL = ws + 11 * SZ;
  float* H = ws + 12 * SZ;  // 16*101*128

  const int M = BB * TT;  // 8000
  const dim3 blk(256);
  auto gblk = [](int tiles) { return dim3((tiles + 7) / 8); };
  const int tiles8000 = (M / 16) * (DK / 16);        // 4000
  const int tiles2000 = (NGLOB / 16) * (DK / 16);    // 1000
  const int nTilesQ = (TT + 15) / 16;                // 32
  const dim3 attnGrid((BB * HH * nTilesQ + 7) / 8);  // 512

  // 1. all0 / save_all (fused gather + W_all GEMM)
  gemm_all0_wmma<<<gblk(tiles8000), blk, 0, stream>>>(
      E_e, E_c, e_data, c_data, a_data, e_diff, W_all, b_all, save);
  // 2. X = swapaxes(all0) + pos
  build_x_kernel<<<(BB * TT * DK + 255) / 256, blk, 0, stream>>>(save, pos_is,
                                                                 X);
  // 3. is-block projections
  gemm_cat_wmma<<<gblk(tiles8000), blk, 0, stream>>>(X, DK, X, 0, Wq_is, bq_is,
                                                     QQ, M, DK);
  gemm_cat_wmma<<<gblk(tiles8000), blk, 0, stream>>>(X, DK, X, 0, Wk_is, bk_is,
                                                     KQ, M, DK);
  gemm_cat_wmma<<<gblk(tiles8000), blk, 0, stream>>>(X, DK, X, 0, Wv_is, bv_is,
                                                     VQ, M, DK);
  gemm_cat_wmma<<<gblk(tiles8000), blk, 0, stream>>>(QQ, DK, QQ, 0, is_wq,
                                                     is_bq, Q2, M, DK);
  gemm_cat_wmma<<<gblk(tiles8000), blk, 0, stream>>>(KQ, DK, KQ, 0, is_wk,
                                                     is_bk, K2, M, DK);
  gemm_cat_wmma<<<gblk(tiles8000), blk, 0, stream>>>(VQ, DK, VQ, 0, is_wv,
                                                     is_bv, V2, M, DK);
  // 4. causal self-attention (K/V rows: k*B*128 + b*128)
  attn_wmma_kernel<<<attnGrid, blk, 0, stream>>>(Q2, K2, V2, CTX, TT, TT,
                                                 BB * DK, DK, nTilesQ);
  // 5. output proj + residual-LN
  gemm_cat_wmma<<<gblk(tiles8000), blk, 0, stream>>>(CTX, DK, CTX, 0, is_wo,
                                                     is_bo, MHO, M, DK);
  add_ln_kernel<<<(M + 7) / 8, blk, 0, stream>>>(MHO, QQ, g_is, be_is, AIS, M);
  // 6. cs-block: kh/vh are batch-independent -> compute per-center (2000 rows)
  gemm_cat_wmma<<<gblk(tiles8000), blk, 0, stream>>>(X, DK, X, 0, Wq_cs, bq_cs,
                                                     QQ, M, DK);  // QH
  gemm_cat_wmma<<<gblk(tiles2000), blk, 0, stream>>>(MC, DK, MC, 0, Wk_cs,
                                                     bk_cs, KQ, NGLOB, DK);
  gemm_cat_wmma<<<gblk(tiles2000), blk, 0, stream>>>(MC, DK, MC, 0, Wv_cs,
                                                     bv_cs, VQ, NGLOB, DK);
  gemm_cat_wmma<<<gblk(tiles8000), blk, 0, stream>>>(QQ, DK, QQ, 0, cs_wq,
                                                     cs_bq, Q2, M, DK);
  gemm_cat_wmma<<<gblk(tiles2000), blk, 0, stream>>>(KQ, DK, KQ, 0, cs_wk,
                                                     cs_bk, K2, NGLOB, DK);
  gemm_cat_wmma<<<gblk(tiles2000), blk, 0, stream>>>(VQ, DK, VQ, 0, cs_wv,
                                                     cs_bv, V2, NGLOB, DK);
  // 7. cross-attention to centers (K/V rows: k*128, batch stride 0)
  attn_wmma_kernel<<<attnGrid, blk, 0, stream>>>(Q2, K2, V2, CTX, TT, NGLOB,
                                                 DK, 0, nTilesQ);
  gemm_cat_wmma<<<gblk(tiles8000), blk, 0, stream>>>(CTX, DK, CTX, 0, cs_wo,
                                                     cs_bo, MHO, M, DK);
  add_ln_kernel<<<(M + 7) / 8, blk, 0, stream>>>(MHO, QQ, g_cs, be_cs, ACS, M);
  // 8. fusion: all_l = W_fc @ [att_is | att_cs] + b_fc
  gemm_cat_wmma<<<gblk(tiles8000), blk, 0, stream>>>(AIS, DK, ACS, DK, W_fc,
                                                     b_fc, ALL, M, DK);
  // 9. sequential DKVMN scan (persistent single block, WMMA gate GEMMs)
  scan_kernel<<<dim3(1), blk, 0, stream>>>(
      ALL, E_at, E_it, E_e, at_data, it_data, e_data, q_matrix, W_lat, b_lat,
      W_lath, b_lath, W_lit, b_lit, W_lith, b_lith, W_y, b_y, h0, H, pred);
}